// ChannelSelfAttention_7473243095550
// MI455X (gfx1250) — compile-verified
//
#include <hip/hip_runtime.h>

// ChannelSelfAttention, algebraically collapsed (exact fp32 reassociation):
//   ym[b,c] = (scale/H) * ( sum_l x[b,l,c]*w[b,l] + cb[b] )
//   w[b,l]  = sum_h Wq[h,l]*t[b,h],  cb[b] = sum_h bq[h]*t[b,h]
//   t[b,h]  = sum_l Wk[h,l]*u[b,l] + bk[h]*S[b]           <-- WMMA f32 GEMM
//   u[b,l]  = sum_c x[b,l,c]*vs[b,c], S[b] = sum_c vs[b,c]
//   vs[b,c] = sum_l x[b,l,c]*wvsum[l] + bvsum, wvsum[l] = sum_h Wv[h,l]
//   out     = x * ym (broadcast over l)
// Memory-bound: ~2 full passes over x from HBM (x fits in 192MB L2 for re-reads).

constexpr int NB = 32;      // batch
constexpr int NL = 4096;    // sequence length (reduction dim for projections)
constexpr int NC = 256;     // channels
constexpr int NH = 64;      // heads
constexpr int NSEG = 32;    // L-split for two-stage dot products
constexpr int LSEG = NL / NSEG;   // 128
constexpr int KB = 32;      // K-split blocks for the WMMA GEMM (4096/32 = 128 per block)

// workspace offsets (floats)
constexpr size_t OFF_WVSUM = 0;                       // [NL]
constexpr size_t OFF_BVSUM = OFF_WVSUM + NL;          // [1] (+pad)
constexpr size_t OFF_VS    = OFF_BVSUM + 256;         // [NB*NC]
constexpr size_t OFF_PART  = OFF_VS + (size_t)NB*NC;  // [NB*NSEG*NC] (reused by pass 5)
constexpr size_t OFF_U     = OFF_PART + (size_t)NB*NSEG*NC; // [NB*NL]
constexpr size_t OFF_S     = OFF_U + (size_t)NB*NL;   // [NB] (+pad)
constexpr size_t OFF_TPART = OFF_S + 32;              // [KB*NB*NH]
constexpr size_t OFF_T     = OFF_TPART + (size_t)KB*NB*NH; // [NB*NH]
constexpr size_t OFF_W     = OFF_T + (size_t)NB*NH;   // [NB*NL]
constexpr size_t OFF_CB    = OFF_W + (size_t)NB*NL;   // [NB] (+pad)
constexpr size_t OFF_YM    = OFF_CB + 32;             // [NB*NC]

typedef float v2f __attribute__((ext_vector_type(2)));
typedef float v8f __attribute__((ext_vector_type(8)));

// ---------------- K0: wvsum[l] = sum_h Wv[h,l]; bvsum = sum_h bv[h] ----------------
__global__ __launch_bounds__(256) void k0_wvsum(const float* __restrict__ Wkv,
                                                const float* __restrict__ bkv,
                                                float* __restrict__ wvsum,
                                                float* __restrict__ bvsum) {
  const int l = blockIdx.x * 256 + threadIdx.x;
  float s = 0.f;
  #pragma unroll 8
  for (int j = 0; j < NH; ++j) s += Wkv[(size_t)(NH + j) * NL + l];
  wvsum[l] = s;
  if (blockIdx.x == 0 && threadIdx.x == 0) {
    float sb = 0.f;
    for (int j = 0; j < NH; ++j) sb += bkv[NH + j];
    bvsum[0] = sb;
  }
}

// ---- K1/K5: partial_{b,seg,c} = sum_{l in seg} x[b,l,c] * wt[b*wtStride + l] ----
__global__ __launch_bounds__(256) void k_xl_dot_partial(const float* __restrict__ x,
                                                        const float* __restrict__ wt,
                                                        int wtStrideB,
                                                        float* __restrict__ part) {
  const int seg = blockIdx.x, b = blockIdx.y, c = threadIdx.x;
  const float* xp = x + (size_t)b * NL * NC + (size_t)seg * LSEG * NC + c;
  const float* wp = wt + (size_t)b * wtStrideB + (size_t)seg * LSEG;
  float a0 = 0.f, a1 = 0.f, a2 = 0.f, a3 = 0.f;
  for (int i = 0; i < LSEG; i += 4) {
    a0 = fmaf(xp[(size_t)(i + 0) * NC], wp[i + 0], a0);
    a1 = fmaf(xp[(size_t)(i + 1) * NC], wp[i + 1], a1);
    a2 = fmaf(xp[(size_t)(i + 2) * NC], wp[i + 2], a2);
    a3 = fmaf(xp[(size_t)(i + 3) * NC], wp[i + 3], a3);
  }
  part[((size_t)b * NSEG + seg) * NC + c] = (a0 + a1) + (a2 + a3);
}

// ---------------- K1b: vs[b,c] = sum_seg part + bvsum; S[b] = sum_c vs[b,c] ----------------
__global__ __launch_bounds__(256) void k1b_vs_reduce(const float* __restrict__ part,
                                                     const float* __restrict__ bvsum,
                                                     float* __restrict__ vs,
                                                     float* __restrict__ S) {
  const int b = blockIdx.x, c = threadIdx.x;
  float s = 0.f;
  #pragma unroll 8
  for (int seg = 0; seg < NSEG; ++seg) s += part[((size_t)b * NSEG + seg) * NC + c];
  const float v = s + bvsum[0];
  vs[(size_t)b * NC + c] = v;
  __shared__ float sh[256];
  sh[c] = v;
  __syncthreads();
  for (int off = 128; off > 0; off >>= 1) {
    if (c < off) sh[c] += sh[c + off];
    __syncthreads();
  }
  if (c == 0) S[b] = sh[0];
}

// ---------------- K2: u[b,l] = dot(x[b,l,:], vs[b,:])  (one wave per l) ----------------
__global__ __launch_bounds__(256) void k2_u(const float* __restrict__ x,
                                            const float* __restrict__ vs,
                                            float* __restrict__ u) {
  const int b = blockIdx.y;
  const int lane = threadIdx.x & 31;
  const int lbase = blockIdx.x * 256 + (threadIdx.x >> 5) * 32;
  const float4* vsp = (const float4*)(vs + (size_t)b * NC);
  const float4 v0 = vsp[lane * 2 + 0];
  const float4 v1 = vsp[lane * 2 + 1];
  for (int i = 0; i < 32; ++i) {
    const int l = lbase + i;
    const float4* xp = (const float4*)(x + (size_t)b * NL * NC + (size_t)l * NC + lane * 8);
    const float4 a = xp[0], bb = xp[1];
    float p = a.x * v0.x + a.y * v0.y + a.z * v0.z + a.w * v0.w
            + bb.x * v1.x + bb.y * v1.y + bb.z * v1.z + bb.w * v1.w;
    #pragma unroll
    for (int off = 16; off > 0; off >>= 1) p += __shfl_xor(p, off, 32);
    if (lane == 0) u[(size_t)b * NL + l] = p;
  }
}

// ---------------- K3: t_part[kb] = u(32xK) @ Wk^T(Kx64) via V_WMMA_F32_16X16X4_F32 ----------------
// A (16x4 f32): lanes 0-15 hold rows M=0..15 with K={0,1} in VGPR{0,1}; lanes 16-31: K={2,3}.
// B (4x16 f32): rows striped across lanes within a VGPR: VGPR0 = K{0 | 2}, VGPR1 = K{1 | 3}
//               (low lanes K=0/1, high lanes K=2/3), N = lane&15.
// D (16x16 f32): VGPR r: lanes 0-15 -> (M=r, N=lane), lanes 16-31 -> (M=8+r, N=lane-16).
__global__ __launch_bounds__(256) void k3_t_partial(const float* __restrict__ u,
                                                    const float* __restrict__ Wkv,
                                                    float* __restrict__ tpart) {
  const int kb   = blockIdx.x;          // K slice of 128
  const int wave = threadIdx.x >> 5;    // 0..7
  const int lane = threadIdx.x & 31;
  const int lo   = lane & 15;
  const int hi   = lane >> 4;           // 0 or 1
  const int mtile = wave >> 2;          // 0..1  (batch rows)
  const int ntile = wave & 3;           // 0..3  (head cols)
  const int m = mtile * 16 + lo;        // batch index (A row)
  const int n = ntile * 16 + lo;        // head index  (B col); Wk[h,l] = Wkv[h*NL + l]
  const int k0 = kb * (NL / KB);

  v8f acc = {};
  for (int kk = 0; kk < NL / KB; kk += 4) {
    const int k = k0 + kk + hi * 2;
    v2f a, bm;
    a.x  = u[(size_t)m * NL + k + 0];
    a.y  = u[(size_t)m * NL + k + 1];
    bm.x = Wkv[(size_t)n * NL + k + 0];   // B[k][n] = Wk[n,k]
    bm.y = Wkv[(size_t)n * NL + k + 1];
    acc = __builtin_amdgcn_wmma_f32_16x16x4_f32(false, a, false, bm,
                                                (short)0, acc, false, false);
  }
  #pragma unroll
  for (int r = 0; r < 8; ++r) {
    const int row = mtile * 16 + hi * 8 + r;    // batch
    const int col = ntile * 16 + lo;            // head
    tpart[((size_t)kb * NB + row) * NH + col] = acc[r];
  }
}

// ---------------- K3b: t[b,h] = sum_kb tpart + bk[h]*S[b]; cb[b] = sum_h bq[h]*t[b,h] ----------------
__global__ __launch_bounds__(64) void k3b_t(const float* __restrict__ tpart,
                                            const float* __restrict__ bkv,
                                            const float* __restrict__ S,
                                            const float* __restrict__ bq,
                                            float* __restrict__ t,
                                            float* __restrict__ cb) {
  const int b = blockIdx.x, h = threadIdx.x;
  float s = 0.f;
  #pragma unroll 8
  for (int kb = 0; kb < KB; ++kb) s += tpart[((size_t)kb * NB + b) * NH + h];
  const float tv = s + bkv[h] * S[b];
  t[(size_t)b * NH + h] = tv;
  __shared__ float sh[64];
  sh[h] = bq[h] * tv;
  __syncthreads();
  for (int off = 32; off > 0; off >>= 1) {
    if (h < off) sh[h] += sh[h + off];
    __syncthreads();
  }
  if (h == 0) cb[b] = sh[0];
}

// ---------------- K4: w[b,l] = sum_h Wq[h,l] * t[b,h] ----------------
__global__ __launch_bounds__(256) void k4_w(const float* __restrict__ Wq,
                                            const float* __restrict__ t,
                                            float* __restrict__ w) {
  const int b = blockIdx.y;
  const int l = blockIdx.x * 256 + threadIdx.x;
  __shared__ float ts[NH];
  if (threadIdx.x < NH) ts[threadIdx.x] = t[(size_t)b * NH + threadIdx.x];
  __syncthreads();
  float acc = 0.f;
  #pragma unroll 8
  for (int h = 0; h < NH; ++h) acc = fmaf(Wq[(size_t)h * NL + l], ts[h], acc);
  w[(size_t)b * NL + l] = acc;
}

// ---------------- K5b: ym[b,c] = (scale/H) * (sum_seg part + cb[b]) ----------------
__global__ __launch_bounds__(256) void k5b_ym_reduce(const float* __restrict__ part,
                                                     const float* __restrict__ cb,
                                                     float* __restrict__ ym) {
  const int b = blockIdx.x, c = threadIdx.x;
  float s = 0.f;
  #pragma unroll 8
  for (int seg = 0; seg < NSEG; ++seg) s += part[((size_t)b * NSEG + seg) * NC + c];
  // scale/H = (H^-0.5)/H = 0.125/64
  ym[(size_t)b * NC + c] = 0.001953125f * (s + cb[b]);
}

// ---------------- K6: out[b,l,c] = x[b,l,c] * ym[b,c]  (pure bandwidth, float4) ----------------
__global__ __launch_bounds__(256) void k6_out(const float* __restrict__ x,
                                              const float* __restrict__ ym,
                                              float* __restrict__ out) {
  const size_t i4 = (size_t)blockIdx.x * 256 + threadIdx.x;
  const size_t base = i4 * 4;
  const int b = (int)(base >> 20);          // L*C = 2^20
  const int c = (int)(base & (NC - 1));     // consecutive c, same (b,l)
  const float4 xv = ((const float4*)x)[i4];
  const float4 yv = *(const float4*)(ym + (size_t)b * NC + c);
  ((float4*)out)[i4] = make_float4(xv.x * yv.x, xv.y * yv.y, xv.z * yv.z, xv.w * yv.w);
}

extern "C" void kernel_launch(void* const* d_in, const int* in_sizes, int n_in,
                              void* d_out, int out_size, void* d_ws, size_t ws_size,
                              hipStream_t stream) {
  const float* x   = (const float*)d_in[0];
  const float* Wq  = (const float*)d_in[1];
  const float* bq  = (const float*)d_in[2];
  const float* Wkv = (const float*)d_in[3];
  const float* bkv = (const float*)d_in[4];
  float* out = (float*)d_out;
  float* ws  = (float*)d_ws;

  float* wvsum = ws + OFF_WVSUM;
  float* bvsum = ws + OFF_BVSUM;
  float* vs    = ws + OFF_VS;
  float* part  = ws + OFF_PART;
  float* u     = ws + OFF_U;
  float* S     = ws + OFF_S;
  float* tpart = ws + OFF_TPART;
  float* t     = ws + OFF_T;
  float* w     = ws + OFF_W;
  float* cb    = ws + OFF_CB;
  float* ym    = ws + OFF_YM;

  // 0) column-sum of Wv and bv
  k0_wvsum<<<NL / 256, 256, 0, stream>>>(Wkv, bkv, wvsum, bvsum);
  // 1) vs[b,c] = x^T wvsum + bvsum   (pass 1 over x, two-stage)
  k_xl_dot_partial<<<dim3(NSEG, NB), 256, 0, stream>>>(x, wvsum, 0, part);
  k1b_vs_reduce<<<NB, 256, 0, stream>>>(part, bvsum, vs, S);
  // 2) u[b,l] = x[b,l,:] . vs[b,:]    (pass 2 over x, L2-resident)
  k2_u<<<dim3(NL / 256, NB), 256, 0, stream>>>(x, vs, u);
  // 3) t = u @ Wk^T (+ bk*S)          (fp32 WMMA GEMM, K-split)
  k3_t_partial<<<KB, 256, 0, stream>>>(u, Wkv, tpart);
  k3b_t<<<NB, 64, 0, stream>>>(tpart, bkv, S, bq, t, cb);
  // 4) w[b,l] = Wq^T t[b]
  k4_w<<<dim3(NL / 256, NB), 256, 0, stream>>>(Wq, t, w);
  // 5) ym[b,c] = (scale/H)(x^T w + cb)  (pass 3 over x, two-stage; reuse `part`)
  k_xl_dot_partial<<<dim3(NSEG, NB), 256, 0, stream>>>(x, w, NL, part);
  k5b_ym_reduce<<<NB, 256, 0, stream>>>(part, cb, ym);
  // 6) out = x * ym                    (pass 4 over x + 1 full write)
  k6_out<<<(size_t)NB * NL * NC / 1024, 256, 0, stream>>>(x, ym, out);
}